// PQCModel_80444737454652
// MI455X (gfx1250) — compile-verified
//
#include <hip/hip_runtime.h>

// MI455X / gfx1250, wave32. Fused kernel:
//  Stage 1: h = tanh(x @ W^T) * scale via V_WMMA_F32_16X16X4_F32 (exact fp32 WMMA).
//           HBM-bound: 192MB read of x -> ~8.3us floor at 23.3 TB/s.
//  Stage 2: 8-qubit real-amplitude circuit sim; 256 amplitudes per sample in
//           8 VGPRs x 32 lanes (amp index bits[4:0]=lane, bits[7:5]=reg).
//           All cross-lane butterflies use DPP/permlanex16 (VALU pipe), NOT the
//           DS pipe, so the sim co-issues with the memory stream instead of
//           serializing on 2 LDS pipes/WGP.

typedef __attribute__((ext_vector_type(2))) float v2f;
typedef __attribute__((ext_vector_type(8))) float v8f;

#define NQ 8
#define SBERT 768
#define WPB 8     // waves per block
#define SPW 16    // samples per wave (one WMMA M-tile)
#define BT 256    // threads per block

// DPP8 selector for lane^4 within groups of 8: [4,5,6,7,0,1,2,3]
#define DPP8_XOR4 ((4<<0)|(5<<3)|(6<<6)|(7<<9)|(0<<12)|(1<<15)|(2<<18)|(3<<21))

// ---- lane^M exchange on the VALU pipe (DPP / permlanex16), DS-free ----------
template <int M>
__device__ __forceinline__ float lane_xor(float v) {
  int i = __float_as_int(v);
  int r;
  if constexpr (M == 1) {
#if __has_builtin(__builtin_amdgcn_update_dpp)
    r = __builtin_amdgcn_update_dpp(i, i, 0xB1, 0xF, 0xF, false);   // quad_perm [1,0,3,2]
#else
    r = __shfl_xor(i, 1, 32);
#endif
  } else if constexpr (M == 2) {
#if __has_builtin(__builtin_amdgcn_update_dpp)
    r = __builtin_amdgcn_update_dpp(i, i, 0x4E, 0xF, 0xF, false);   // quad_perm [2,3,0,1]
#else
    r = __shfl_xor(i, 2, 32);
#endif
  } else if constexpr (M == 4) {
#if __has_builtin(__builtin_amdgcn_mov_dpp8)
    r = __builtin_amdgcn_mov_dpp8(i, DPP8_XOR4);                    // dpp8 [4,5,6,7,0,1,2,3]
#else
    r = __shfl_xor(i, 4, 32);
#endif
  } else if constexpr (M == 8) {
#if __has_builtin(__builtin_amdgcn_update_dpp)
    r = __builtin_amdgcn_update_dpp(i, i, 0x128, 0xF, 0xF, false);  // row_ror:8 == xor8
#else
    r = __shfl_xor(i, 8, 32);
#endif
  } else {  // M == 16
#if __has_builtin(__builtin_amdgcn_permlanex16)
    r = __builtin_amdgcn_permlanex16(i, i, 0x76543210, 0xFEDCBA98, false, false);
#else
    r = __shfl_xor(i, 16, 32);
#endif
  }
  return __int_as_float(r);
}

// RY on a register-bit pair: a0' = c*a0 - s*a1 ; a1' = s*a0 + c*a1
#define RY_REG(A, i0, i1, C, S)                                  \
  { float _a0 = A[i0], _a1 = A[i1];                              \
    A[i0] = (C) * _a0 - (S) * _a1;                               \
    A[i1] = (S) * _a0 + (C) * _a1; }

// RY on a lane-bit pair: v' = c*v + (bit? +s : -s)*partner
template <int M>
__device__ __forceinline__ void ry_lane(float amp[8], int lane, float c, float s) {
  float sg = (lane & M) ? s : -s;
#pragma unroll
  for (int r = 0; r < 8; ++r) {
    float p = lane_xor<M>(amp[r]);
    amp[r] = c * amp[r] + sg * p;
  }
}

// CNOT: control = lane bit SELM, target = lane bit SHFLM
template <int SHFLM, int SELM>
__device__ __forceinline__ void cnot_sel(float amp[8], int lane) {
#pragma unroll
  for (int r = 0; r < 8; ++r) {
    float p = lane_xor<SHFLM>(amp[r]);
    amp[r] = (lane & SELM) ? p : amp[r];
  }
}

__global__ __launch_bounds__(BT) void pqc_fused_kernel(
    const float* __restrict__ x,       // (B, 768)
    const float* __restrict__ Wp,      // (8, 768)
    const float* __restrict__ scalep,  // scalar
    const float* __restrict__ wts,     // (2, 8)
    float* __restrict__ out) {         // (B, 8)

  __shared__ float  sW[16 * SBERT];          // W padded to 16 rows (rows 8..15 = 0), 48KB
  __shared__ float2 sCS[WPB][SPW][NQ];       // (cos, sin) of half-angles, 8KB

  const int tid  = threadIdx.x;
  const int lane = tid & 31;
  const int wave = tid >> 5;
  const int half = lane >> 4;     // K half-select for WMMA fragments
  const int lrow = lane & 15;     // M (A) / N (B) index

  // ---- stage W into LDS, zero-pad rows 8..15 (kills divergence in the K loop)
  {
    const float4* src = (const float4*)Wp;
    float4* dst = (float4*)sW;
    const float4 z4 = make_float4(0.f, 0.f, 0.f, 0.f);
    for (int i = tid; i < (16 * SBERT) / 4; i += BT)
      dst[i] = (i < (NQ * SBERT) / 4) ? src[i] : z4;
  }
  __syncthreads();

  const size_t sampleBase = ((size_t)blockIdx.x * WPB + wave) * SPW;
  const float* xr = x + (sampleBase + (size_t)lrow) * SBERT;
  const float* wr = sW + lrow * SBERT;

  // ---- projection GEMM: D(16x16) += A(16x4) * B(4x16), K = 768 in steps of 4
  v8f acc = {};
#pragma unroll 8
  for (int k = 0; k < SBERT; k += 4) {
    const int kk = k + 2 * half;
    v2f a = *(const v2f*)(xr + kk);   // A: lane m=lrow, K = kk..kk+1
    v2f b = *(const v2f*)(wr + kk);   // B: lane n=lrow, K = kk..kk+1
    acc = __builtin_amdgcn_wmma_f32_16x16x4_f32(
        false, a, false, b, (short)0, acc, false, false);
  }

  // ---- theta = tanh(dot)*scale ; store (cos(theta/2), sin(theta/2)) per (sample,qubit)
  if (lrow < NQ) {
    const float scaleV = *scalep;
#pragma unroll
    for (int r = 0; r < 8; ++r) {
      float th = 0.5f * scaleV * tanhf(acc[r]);
      sCS[wave][r + 8 * half][lrow] = make_float2(cosf(th), sinf(th));
    }
  }
  __syncthreads();

  // ---- variational half-angle cos/sin (uniform across samples)
  float cw[2][8], sw2[2][8];
#pragma unroll
  for (int L = 0; L < 2; ++L)
#pragma unroll
    for (int q = 0; q < 8; ++q) {
      float wv = 0.5f * wts[L * 8 + q];
      cw[L][q]  = cosf(wv);
      sw2[L][q] = sinf(wv);
    }

  // ---- simulate 16 samples per wave. amp index a: bits[4:0]=lane, bits[7:5]=reg.
  // wire w acts on bit (7-w): wires 0..2 -> reg bits 2..0, wires 3..7 -> lane bits 4..0.
#pragma unroll 1
  for (int s = 0; s < SPW; ++s) {
    float ce[8], se[8];
#pragma unroll
    for (int q = 0; q < 8; ++q) {
      float2 t = sCS[wave][s][q];
      ce[q] = t.x; se[q] = t.y;
    }

    // encoding layer applied analytically: product state
    float lf = 1.f;
    lf *= (lane & 1)  ? se[7] : ce[7];
    lf *= (lane & 2)  ? se[6] : ce[6];
    lf *= (lane & 4)  ? se[5] : ce[5];
    lf *= (lane & 8)  ? se[4] : ce[4];
    lf *= (lane & 16) ? se[3] : ce[3];
    float amp[8];
#pragma unroll
    for (int r = 0; r < 8; ++r) {
      float rf = ((r & 1) ? se[2] : ce[2]);
      rf      *= ((r & 2) ? se[1] : ce[1]);
      rf      *= ((r & 4) ? se[0] : ce[0]);
      amp[r] = lf * rf;
    }

#pragma unroll
    for (int L = 0; L < 2; ++L) {
      // RY wire0 (reg bit2)
      RY_REG(amp, 0, 4, cw[L][0], sw2[L][0]); RY_REG(amp, 1, 5, cw[L][0], sw2[L][0]);
      RY_REG(amp, 2, 6, cw[L][0], sw2[L][0]); RY_REG(amp, 3, 7, cw[L][0], sw2[L][0]);
      // RY wire1 (reg bit1)
      RY_REG(amp, 0, 2, cw[L][1], sw2[L][1]); RY_REG(amp, 1, 3, cw[L][1], sw2[L][1]);
      RY_REG(amp, 4, 6, cw[L][1], sw2[L][1]); RY_REG(amp, 5, 7, cw[L][1], sw2[L][1]);
      // RY wire2 (reg bit0)
      RY_REG(amp, 0, 1, cw[L][2], sw2[L][2]); RY_REG(amp, 2, 3, cw[L][2], sw2[L][2]);
      RY_REG(amp, 4, 5, cw[L][2], sw2[L][2]); RY_REG(amp, 6, 7, cw[L][2], sw2[L][2]);
      // RY wires 3..7 (lane bits 4..0) -- DPP/permlane butterflies
      ry_lane<16>(amp, lane, cw[L][3], sw2[L][3]);
      ry_lane<8> (amp, lane, cw[L][4], sw2[L][4]);
      ry_lane<4> (amp, lane, cw[L][5], sw2[L][5]);
      ry_lane<2> (amp, lane, cw[L][6], sw2[L][6]);
      ry_lane<1> (amp, lane, cw[L][7], sw2[L][7]);

      // CNOT chain ctrl=i, tgt=i+1  => (cbit,tbit) = (7-i, 6-i)
      { float t = amp[4]; amp[4] = amp[6]; amp[6] = t;     // (7,6): reg bit2 -> flip reg bit1
        t = amp[5]; amp[5] = amp[7]; amp[7] = t; }
      { float t = amp[2]; amp[2] = amp[3]; amp[3] = t;     // (6,5): reg bit1 -> flip reg bit0
        t = amp[6]; amp[6] = amp[7]; amp[7] = t; }
      amp[1] = lane_xor<16>(amp[1]);                       // (5,4): odd regs flip lane bit4
      amp[3] = lane_xor<16>(amp[3]);
      amp[5] = lane_xor<16>(amp[5]);
      amp[7] = lane_xor<16>(amp[7]);
      cnot_sel<8, 16>(amp, lane);                          // (4,3)
      cnot_sel<4, 8> (amp, lane);                          // (3,2)
      cnot_sel<2, 4> (amp, lane);                          // (2,1)
      cnot_sel<1, 2> (amp, lane);                          // (1,0)
    }

    // ---- readout: <Z_w> = sum amp^2 * (1 - 2*bit_{7-w})
    float sq[8];
#pragma unroll
    for (int r = 0; r < 8; ++r) sq[r] = amp[r] * amp[r];
    // reg-bit signs are lane-uniform: plain lane reductions
    float A0 = (sq[0] + sq[1] + sq[2] + sq[3]) - (sq[4] + sq[5] + sq[6] + sq[7]);
    float A1 = (sq[0] + sq[1] + sq[4] + sq[5]) - (sq[2] + sq[3] + sq[6] + sq[7]);
    float A2 = (sq[0] + sq[2] + sq[4] + sq[6]) - (sq[1] + sq[3] + sq[5] + sq[7]);
    float T  = sq[0] + sq[1] + sq[2] + sq[3] + sq[4] + sq[5] + sq[6] + sq[7];
#define RED_SUM(V)                                             \
    V += lane_xor<1>(V);  V += lane_xor<2>(V);                 \
    V += lane_xor<4>(V);  V += lane_xor<8>(V);  V += lane_xor<16>(V);
    RED_SUM(A0); RED_SUM(A1); RED_SUM(A2);
#undef RED_SUM
    // Walsh-Hadamard butterfly on T: lane (1<<b) ends with sum_l (-1)^{bit_b(l)} T_l,
    // giving z3..z7 (lane-bit signed reductions) in one 5-stage pass.
    { float p = lane_xor<1>(T);  T = (lane & 1)  ? p - T : p + T; }
    { float p = lane_xor<2>(T);  T = (lane & 2)  ? p - T : p + T; }
    { float p = lane_xor<4>(T);  T = (lane & 4)  ? p - T : p + T; }
    { float p = lane_xor<8>(T);  T = (lane & 8)  ? p - T : p + T; }
    { float p = lane_xor<16>(T); T = (lane & 16) ? p - T : p + T; }
    int ti = __float_as_int(T);
    float z3 = __int_as_float(__builtin_amdgcn_readlane(ti, 16));
    float z4 = __int_as_float(__builtin_amdgcn_readlane(ti, 8));
    float z5 = __int_as_float(__builtin_amdgcn_readlane(ti, 4));
    float z6 = __int_as_float(__builtin_amdgcn_readlane(ti, 2));
    float z7 = __int_as_float(__builtin_amdgcn_readlane(ti, 1));

    if (lane == 0) {
      float4* o = (float4*)(out + (sampleBase + (size_t)s) * NQ);
      o[0] = make_float4(A0, A1, A2, z3);
      o[1] = make_float4(z4, z5, z6, z7);
    }
  }
}

extern "C" void kernel_launch(void* const* d_in, const int* in_sizes, int n_in,
                              void* d_out, int out_size, void* d_ws, size_t ws_size,
                              hipStream_t stream) {
  const float* x       = (const float*)d_in[0];
  const float* W_proj  = (const float*)d_in[1];
  const float* scalep  = (const float*)d_in[2];
  const float* weights = (const float*)d_in[3];
  float* out = (float*)d_out;

  const int B = in_sizes[0] / SBERT;               // 65536
  const int grid = B / (WPB * SPW);                // 512 blocks, 16 samples/wave
  pqc_fused_kernel<<<grid, BT, 0, stream>>>(x, W_proj, scalep, weights, out);
}